// SorecModel_42374147343140
// MI455X (gfx1250) — compile-verified
//
#include <hip/hip_runtime.h>
#include <math.h>

// ---------------------------------------------------------------------------
// SoRec model on MI455X (gfx1250): edge-dot via V_WMMA_F32_16X16X4_F32.
// One wave handles 16 edges: A = 16 gathered src rows (16x64), B = 16 gathered
// dst rows as 64x16; diag(A x B^T) = the 16 edge dot products. Bandwidth is
// identical to a scalar dot (each lane loads contiguous float2 fragments),
// compute waste is irrelevant (workload is L2-gather bound, tables fit in the
// 192MB L2).
// ---------------------------------------------------------------------------

typedef __attribute__((ext_vector_type(2))) float v2f;
typedef __attribute__((ext_vector_type(8))) float v8f;

#define LAMDA_F 0.5f

// ---------------- utility kernels ----------------

__global__ void zero_ws_kernel(float* __restrict__ p, int n) {
  int i = blockIdx.x * blockDim.x + threadIdx.x;
  if (i < n) p[i] = 0.0f;
}

__global__ void degree_kernel(const int* __restrict__ src, const int* __restrict__ dst,
                              float* __restrict__ in_deg, float* __restrict__ out_deg,
                              int n) {
  int stride = gridDim.x * blockDim.x;
  for (int i = blockIdx.x * blockDim.x + threadIdx.x; i < n; i += stride) {
    atomicAdd(&out_deg[src[i]], 1.0f);
    atomicAdd(&in_deg[dst[i]], 1.0f);
  }
}

// sum of squares over n4 float4 elements -> atomicAdd into *acc
__global__ void sumsq_kernel(const float* __restrict__ x, int n4,
                             float* __restrict__ acc) {
  float s = 0.0f;
  int stride = gridDim.x * blockDim.x;
  const float4* x4 = (const float4*)x;
  for (int i = blockIdx.x * blockDim.x + threadIdx.x; i < n4; i += stride) {
    float4 v = x4[i];
    s = fmaf(v.x, v.x, s);
    s = fmaf(v.y, v.y, s);
    s = fmaf(v.z, v.z, s);
    s = fmaf(v.w, v.w, s);
  }
#pragma unroll
  for (int off = 16; off > 0; off >>= 1) s += __shfl_xor(s, off, 32);
  if ((threadIdx.x & 31) == 0) atomicAdd(acc, s);
}

// ---------------- WMMA edge-dot core ----------------
//
// f32 WMMA 16x16x4 A-fragment layout (32-bit A 16x4): lane L (L%16 = row M)
// holds K = 2*(L/16) in VGPR0 and K = 2*(L/16)+1 in VGPR1 -> a contiguous
// float2 at h_src[row*64 + k + 2*(L/16)]. B (4x16, K-major across VGPRs,
// N across lanes) is the mirror image with row -> dst column, i.e. the same
// contiguous float2 from the dst row (B = dst_rows^T).
__device__ __forceinline__ v8f edge_tile_mma(const float* __restrict__ hsrc,
                                             const float* __restrict__ hdst,
                                             const int* __restrict__ src,
                                             const int* __restrict__ dst,
                                             long base, int nE, int lane) {
  int e = lane & 15;               // edge (row of A / column of B) for this lane
  int kofs = (lane >> 4) << 1;     // 0 for lanes 0-15, 2 for lanes 16-31
  long eidx = base + e;
  if (eidx >= nE) eidx = nE - 1;   // keep EXEC all-ones; tail lanes clamp
  const float* pa = hsrc + (long)src[eidx] * 64 + kofs;
  const float* pb = hdst + (long)dst[eidx] * 64 + kofs;

  v8f acc = {};
#pragma unroll
  for (int k = 0; k < 64; k += 4) {
    v2f a = *(const v2f*)(pa + k);
    v2f b = *(const v2f*)(pb + k);
    acc = __builtin_amdgcn_wmma_f32_16x16x4_f32(
        /*neg_a=*/false, a, /*neg_b=*/false, b,
        /*c_mod=*/(short)0, acc, /*reuse_a=*/false, /*reuse_b=*/false);
  }
  return acc;
}

// C/D 16x16 f32 layout: lanes 0-15 hold (M=v, N=lane), lanes 16-31 hold
// (M=v+8, N=lane-16). Diagonal (e,e): e<8 -> lane e, VGPR e; e>=8 ->
// lane e+16, VGPR e-8. Returns the diagonal value; *out_e = edge-in-tile
// index owned by this lane, or -1.
__device__ __forceinline__ float diag_of(v8f acc, int lane, int* out_e) {
  int e = -1, v = 0;
  if (lane < 8) {
    e = lane;
    v = lane;
  } else if (lane >= 24) {
    e = lane - 16;
    v = lane - 24;
  }
  float d = acc[0];
#pragma unroll
  for (int i = 1; i < 8; ++i) d = (v == i) ? acc[i] : d;
  *out_e = e;
  return d;
}

// ---------------- score kernel (pos / neg) ----------------

__global__ void edge_score_wmma(const float* __restrict__ hsrc,
                                const float* __restrict__ hdst,
                                const int* __restrict__ src,
                                const int* __restrict__ dst,
                                float* __restrict__ out, int nE, int nTiles) {
  int wave = (blockIdx.x * blockDim.x + threadIdx.x) >> 5;  // one 16-edge tile per wave
  int lane = threadIdx.x & 31;
  if (wave >= nTiles) return;  // wave-uniform exit
  long base = (long)wave * 16;

  v8f acc = edge_tile_mma(hsrc, hdst, src, dst, base, nE, lane);

  int e;
  float d = diag_of(acc, lane, &e);
  if (e >= 0) {
    long o = base + e;
    if (o < nE) out[o] = d;
  }
}

// ---------------- link-loss kernel ----------------

__global__ void link_loss_wmma(const float* __restrict__ hsrc,   // y_w_user
                               const float* __restrict__ hdst,   // p_q_user
                               const int* __restrict__ src,
                               const int* __restrict__ dst,
                               const float* __restrict__ in_deg,
                               const float* __restrict__ out_deg,
                               float* __restrict__ loss_acc, int nE, int nTiles) {
  int wave = (blockIdx.x * blockDim.x + threadIdx.x) >> 5;
  int lane = threadIdx.x & 31;
  if (wave >= nTiles) return;
  long base = (long)wave * 16;

  v8f acc = edge_tile_mma(hsrc, hdst, src, dst, base, nE, lane);

  int e;
  float d = diag_of(acc, lane, &e);

  float contrib = 0.0f;
  if (e >= 0) {
    long o = base + e;
    if (o < nE) {
      float pred = 1.0f / (1.0f + __expf(-d));
      float din = in_deg[dst[o]];
      float dout = out_deg[src[o]];
      float label = sqrtf(din / (dout + din));
      float diff = pred - label;
      contrib = diff * diff;
    }
  }
#pragma unroll
  for (int off = 16; off > 0; off >>= 1) contrib += __shfl_xor(contrib, off, 32);
  if (lane == 0) atomicAdd(loss_acc, contrib);
}

// ---------------- finalize ----------------

__global__ void finalize_kernel(const float* __restrict__ accs,
                                float* __restrict__ reg_out,
                                float* __restrict__ link_out, float invE) {
  reg_out[0] = LAMDA_F * (sqrtf(accs[0]) + sqrtf(accs[1]) + sqrtf(accs[2]));
  link_out[0] = accs[3] * invE;
}

// ---------------------------------------------------------------------------

extern "C" void kernel_launch(void* const* d_in, const int* in_sizes, int n_in,
                              void* d_out, int out_size, void* d_ws, size_t ws_size,
                              hipStream_t stream) {
  const float* p_q_user = (const float*)d_in[0];
  const float* p_q_item = (const float*)d_in[1];
  const float* y_w_user = (const float*)d_in[2];
  // d_in[3] = y_w_item (unused by the reference outputs)
  const int* t_src = (const int*)d_in[4];
  const int* t_dst = (const int*)d_in[5];
  const int* p_src = (const int*)d_in[6];
  const int* p_dst = (const int*)d_in[7];
  const int* n_src = (const int*)d_in[8];
  const int* n_dst = (const int*)d_in[9];

  const int U = in_sizes[0] / 64;   // 100000
  const int E_T = in_sizes[4];      // 1M
  const int E_P = in_sizes[6];      // 1M
  const int E_N = in_sizes[8];      // 1M

  // workspace: [0,U) in_deg | [U,2U) out_deg | [2U..2U+3) sumsq accs | [2U+3] link acc
  float* ws = (float*)d_ws;
  float* in_deg = ws;
  float* out_deg = ws + U;
  float* accs = ws + 2 * (long)U;

  float* out = (float*)d_out;
  float* pos_out = out;
  float* neg_out = out + E_P;
  float* reg_out = out + (long)E_P + E_N;
  float* link_out = reg_out + 1;

  const int zn = 2 * U + 4;
  zero_ws_kernel<<<(zn + 255) / 256, 256, 0, stream>>>(ws, zn);

  degree_kernel<<<2048, 256, 0, stream>>>(t_src, t_dst, in_deg, out_deg, E_T);

  sumsq_kernel<<<2048, 256, 0, stream>>>(p_q_user, in_sizes[0] / 4, accs + 0);
  sumsq_kernel<<<2048, 256, 0, stream>>>(p_q_item, in_sizes[1] / 4, accs + 1);
  sumsq_kernel<<<2048, 256, 0, stream>>>(y_w_user, in_sizes[2] / 4, accs + 2);

  const int tilesP = (E_P + 15) / 16;
  const int tilesN = (E_N + 15) / 16;
  const int tilesT = (E_T + 15) / 16;
  const int wavesPerBlock = 256 / 32;

  edge_score_wmma<<<(tilesP + wavesPerBlock - 1) / wavesPerBlock, 256, 0, stream>>>(
      p_q_user, p_q_item, p_src, p_dst, pos_out, E_P, tilesP);
  edge_score_wmma<<<(tilesN + wavesPerBlock - 1) / wavesPerBlock, 256, 0, stream>>>(
      p_q_user, p_q_item, n_src, n_dst, neg_out, E_N, tilesN);

  link_loss_wmma<<<(tilesT + wavesPerBlock - 1) / wavesPerBlock, 256, 0, stream>>>(
      y_w_user, p_q_user, t_src, t_dst, in_deg, out_deg, accs + 3, E_T, tilesT);

  finalize_kernel<<<1, 1, 0, stream>>>(accs, reg_out, link_out, 1.0f / (float)E_T);
}